// Threshold_weights5_52699248721949
// MI455X (gfx1250) — compile-verified
//
#include <hip/hip_runtime.h>
#include <math.h>

#define WAVES_PER_BLOCK 8
#define ROWS_PER_WAVE   4
#define ROWS_PER_BLOCK  (WAVES_PER_BLOCK * ROWS_PER_WAVE)
#define NCOL   100
#define CPAD   104   // padded row length (floats)
#define NMAT   6

typedef int v4i __attribute__((ext_vector_type(4)));
typedef __attribute__((address_space(1))) v4i as1_v4i;
typedef __attribute__((address_space(3))) v4i as3_v4i;

// ---- CDNA5 async global -> LDS copy (ASYNCcnt-tracked) -------------------
#if __has_builtin(__builtin_amdgcn_global_load_async_to_lds_b128)
#define ASYNC_CP16(gsrc, ldst)                                              \
  __builtin_amdgcn_global_load_async_to_lds_b128(                           \
      (as1_v4i*)(unsigned long long)(gsrc),                                 \
      (as3_v4i*)(unsigned)(unsigned long long)(ldst), 0, 0)
#else
#define ASYNC_CP16(gsrc, ldst)                                              \
  do {                                                                      \
    unsigned _loff = (unsigned)(unsigned long long)(ldst);                  \
    asm volatile("global_load_async_to_lds_b128 %0, %1, off"                \
                 :: "v"(_loff), "v"(gsrc) : "memory");                      \
  } while (0)
#endif

#if __has_builtin(__builtin_amdgcn_s_wait_asynccnt)
#define WAIT_ASYNC(n) do { __builtin_amdgcn_s_wait_asynccnt(n); \
                           asm volatile("" ::: "memory"); } while (0)
#else
#define WAIT_ASYNC(n) asm volatile("s_wait_asynccnt " #n ::: "memory")
#endif

__global__ __launch_bounds__(256)
void margins_softmax_kernel(const float* __restrict__ o1, const float* __restrict__ o2,
                            const float* __restrict__ o3, const float* __restrict__ o4,
                            const float* __restrict__ o5, const float* __restrict__ mm,
                            const int*   __restrict__ targets,
                            float* __restrict__ out,        // [1 + nrows*6]
                            float* __restrict__ blockmax,   // [gridDim.x]
                            int nrows)
{
    __shared__ float buf[WAVES_PER_BLOCK][2][NMAT][CPAD];
    __shared__ float wmax[WAVES_PER_BLOCK];

    const int lane = threadIdx.x & 31;
    const int w    = threadIdx.x >> 5;
    const float* mats[NMAT] = {o1, o2, o3, o4, o5, mm};

    const int base = ((int)blockIdx.x * WAVES_PER_BLOCK + w) * ROWS_PER_WAVE;
    float macc = -__builtin_inff();

    auto issue = [&](int r, int s) {
        if (r < nrows && lane < 25) {
#pragma unroll
            for (int m = 0; m < NMAT; ++m) {
                ASYNC_CP16(mats[m] + (size_t)r * NCOL + lane * 4,
                           &buf[w][s][m][lane * 4]);
            }
        }
    };

    issue(base, 0);  // prime the pipe
#pragma unroll
    for (int i = 0; i < ROWS_PER_WAVE; ++i) {
        const int r = base + i;
        const int s = i & 1;
        const bool hasNext = (i + 1 < ROWS_PER_WAVE) && (base + i + 1 < nrows);
        if (hasNext) issue(base + i + 1, s ^ 1);   // overlap next row's DMA

        if (r < nrows) {
            if (hasNext) { WAIT_ASYNC(6); } else { WAIT_ASYNC(0); }

            const int t = targets[r];
            float margins[NMAT];
#pragma unroll
            for (int m = 0; m < NMAT; ++m) {
                const float* row = &buf[w][s][m][0];
                float a = row[lane];
                float b = row[lane + 32];
                float c = row[lane + 64];
                float d = (lane < 4) ? row[lane + 96] : -__builtin_inff();
                // local top-2 of {a,b,c,d}
                float hi1 = fmaxf(a, b), lo1 = fminf(a, b);
                float hi2 = fmaxf(c, d), lo2 = fminf(c, d);
                float m1 = fmaxf(hi1, hi2);
                float m2 = fmaxf(fminf(hi1, hi2), fmaxf(lo1, lo2));
                // wave32 butterfly top-2 merge
#pragma unroll
                for (int off = 16; off > 0; off >>= 1) {
                    float p1 = __shfl_xor(m1, off, 32);
                    float p2 = __shfl_xor(m2, off, 32);
                    float n1 = fmaxf(m1, p1);
                    float n2 = fmaxf(fminf(m1, p1), fmaxf(m2, p2));
                    m1 = n1; m2 = n2;
                }
                float tv = row[t];                       // broadcast LDS read
                margins[m] = (tv == m1) ? (m1 - m2) : 0.0f;
                if (m < 5) macc = fmaxf(macc, m1);       // global max excludes mimic
            }
            // softmax(margins / 2), numerically stable; uniform across lanes
            float mx = margins[0];
#pragma unroll
            for (int m = 1; m < NMAT; ++m) mx = fmaxf(mx, margins[m]);
            float sum = 0.f;
#pragma unroll
            for (int m = 0; m < NMAT; ++m) sum += expf((margins[m] - mx) * 0.5f);
            float inv = 1.0f / sum;
            // lane j (<6) writes element j
            float mg = margins[0];
#pragma unroll
            for (int m = 1; m < NMAT; ++m) if (lane == m) mg = margins[m];
            if (lane < NMAT)
                out[1 + (size_t)r * NMAT + lane] = expf((mg - mx) * 0.5f) * inv;
        }
    }

    if (lane == 0) wmax[w] = macc;
    __syncthreads();
    if (threadIdx.x == 0) {
        float v = wmax[0];
#pragma unroll
        for (int k = 1; k < WAVES_PER_BLOCK; ++k) v = fmaxf(v, wmax[k]);
        blockmax[blockIdx.x] = v;
    }
}

__global__ __launch_bounds__(256)
void reduce_max_kernel(const float* __restrict__ part, int n, float* __restrict__ out)
{
    __shared__ float s[256];
    float v = -__builtin_inff();
    for (int i = threadIdx.x; i < n; i += 256) v = fmaxf(v, part[i]);
    s[threadIdx.x] = v;
    __syncthreads();
    for (int k = 128; k > 0; k >>= 1) {
        if ((int)threadIdx.x < k) s[threadIdx.x] = fmaxf(s[threadIdx.x], s[threadIdx.x + k]);
        __syncthreads();
    }
    if (threadIdx.x == 0) out[0] = s[0];
}

extern "C" void kernel_launch(void* const* d_in, const int* in_sizes, int n_in,
                              void* d_out, int out_size, void* d_ws, size_t ws_size,
                              hipStream_t stream) {
    const float* o1 = (const float*)d_in[0];
    const float* o2 = (const float*)d_in[1];
    const float* o3 = (const float*)d_in[2];
    const float* o4 = (const float*)d_in[3];
    const float* o5 = (const float*)d_in[4];
    const float* mm = (const float*)d_in[5];
    const int* targets = (const int*)d_in[6];
    const int nrows = in_sizes[6];

    float* out  = (float*)d_out;
    float* part = (float*)d_ws;

    const int grid = (nrows + ROWS_PER_BLOCK - 1) / ROWS_PER_BLOCK;  // 2048 for N=65536
    margins_softmax_kernel<<<grid, 256, 0, stream>>>(o1, o2, o3, o4, o5, mm,
                                                     targets, out, part, nrows);
    reduce_max_kernel<<<1, 256, 0, stream>>>(part, grid, out);
}